// QuantumLayer_88098369176448
// MI455X (gfx1250) — compile-verified
//
#include <hip/hip_runtime.h>
#include <math.h>

#define QN 8
#define QB 256
#define QL 3
#define DIM 65536          // 4^QN : length of vec(rho) per sample
#define SHOTS_F 1024.0f
#define NOISE_P 0.05f

typedef float v2f __attribute__((ext_vector_type(2)));
typedef float v8f __attribute__((ext_vector_type(8)));

struct cplx { float re, im; };
__device__ inline cplx cmul(cplx a, cplx b){ return {a.re*b.re - a.im*b.im, a.re*b.im + a.im*b.re}; }
__device__ inline cplx cadd(cplx a, cplx b){ return {a.re + b.re, a.im + b.im}; }
__device__ inline cplx cconj(cplx a){ return {a.re, -a.im}; }

// ---------------------------------------------------------------------------
// Setup: composed noise superop (4x4 -> 16x16 pair) and CNOT superop (16x16)
// ---------------------------------------------------------------------------
__global__ void k_build_const(float* n16Re, float* n16Im, float* cnRe, float* cnIm)
{
    const float p  = NOISE_P;
    const float s1 = sqrtf(1.0f - p), sp = sqrtf(p), s3 = sqrtf(p / 3.0f);

    cplx K[5][4][2][2];
    int nk[5] = {4, 2, 2, 2, 2};
    for (int c = 0; c < 5; ++c)
        for (int k = 0; k < 4; ++k)
            for (int i = 0; i < 2; ++i)
                for (int j = 0; j < 2; ++j)
                    K[c][k][i][j] = {0.f, 0.f};
    // depolarizing: s1*I, s3*X, s3*Y, s3*Z
    K[0][0][0][0] = {s1,0.f}; K[0][0][1][1] = {s1,0.f};
    K[0][1][0][1] = {s3,0.f}; K[0][1][1][0] = {s3,0.f};
    K[0][2][0][1] = {0.f,-s3}; K[0][2][1][0] = {0.f,s3};
    K[0][3][0][0] = {s3,0.f}; K[0][3][1][1] = {-s3,0.f};
    // bit flip
    K[1][0][0][0] = {s1,0.f}; K[1][0][1][1] = {s1,0.f};
    K[1][1][0][1] = {sp,0.f}; K[1][1][1][0] = {sp,0.f};
    // phase flip
    K[2][0][0][0] = {s1,0.f}; K[2][0][1][1] = {s1,0.f};
    K[2][1][0][0] = {sp,0.f}; K[2][1][1][1] = {-sp,0.f};
    // amplitude damping
    K[3][0][0][0] = {1.f,0.f}; K[3][0][1][1] = {s1,0.f};
    K[3][1][0][1] = {sp,0.f};
    // phase damping
    K[4][0][0][0] = {1.f,0.f}; K[4][0][1][1] = {s1,0.f};
    K[4][1][1][1] = {sp,0.f};

    cplx M[4][4];
    for (int i = 0; i < 4; ++i)
        for (int j = 0; j < 4; ++j)
            M[i][j] = {(i==j) ? 1.f : 0.f, 0.f};

    for (int c = 0; c < 5; ++c) {
        cplx Ch[4][4];
        for (int g = 0; g < 4; ++g)
            for (int gp = 0; gp < 4; ++gp) {
                int r = g >> 1, cc = g & 1, rp = gp >> 1, cp = gp & 1;
                cplx s = {0.f, 0.f};
                for (int k = 0; k < nk[c]; ++k)
                    s = cadd(s, cmul(K[c][k][r][rp], cconj(K[c][k][cc][cp])));
                Ch[g][gp] = s;
            }
        cplx T[4][4];
        for (int i = 0; i < 4; ++i)
            for (int j = 0; j < 4; ++j) {
                cplx s = {0.f, 0.f};
                for (int m2 = 0; m2 < 4; ++m2) s = cadd(s, cmul(Ch[i][m2], M[m2][j]));
                T[i][j] = s;
            }
        for (int i = 0; i < 4; ++i)
            for (int j = 0; j < 4; ++j) M[i][j] = T[i][j];
    }

    const int t = threadIdx.x;            // 256 threads: one 16x16 entry each
    const int R = t >> 4, C = t & 15;
    cplx e = cmul(M[R >> 2][C >> 2], M[R & 3][C & 3]);
    n16Re[t] = e.re; n16Im[t] = e.im;

    // CNOT superop on (control axis, target axis): G = (rc*2+cc)*4 + (rt*2+ct)
    int rc  = (R >> 3) & 1, cc  = (R >> 2) & 1, rt  = (R >> 1) & 1, ct  = R & 1;
    int rcp = (C >> 3) & 1, ccp = (C >> 2) & 1, rtp = (C >> 1) & 1, ctp = C & 1;
    float v = (rc == rcp && cc == ccp && rt == (rtp ^ rcp) && ct == (ctp ^ ccp)) ? 1.f : 0.f;
    cnRe[t] = v; cnIm[t] = 0.f;
}

// ---------------------------------------------------------------------------
// Per-batch RX pair superops, per-layer Rot pair superops
// ---------------------------------------------------------------------------
__device__ inline cplx rx_elem(float a, int i, int j)
{
    if (i == j) return {cosf(0.5f * a), 0.f};
    return {0.f, -sinf(0.5f * a)};
}
__device__ inline cplx superop_rx(float a, int g, int gp)
{
    int r = g >> 1, c = g & 1, rp = gp >> 1, cp = gp & 1;
    return cmul(rx_elem(a, r, rp), cconj(rx_elem(a, c, cp)));
}

__global__ void k_build_rx(const float* __restrict__ x, float* rxRe, float* rxIm)
{
    const int b  = blockIdx.x >> 2;
    const int pr = blockIdx.x & 3;
    const int t  = threadIdx.x;
    const int R = t >> 4, C = t & 15;
    cplx s0 = superop_rx(x[b * QN + 2 * pr + 0], R >> 2, C >> 2);
    cplx s1 = superop_rx(x[b * QN + 2 * pr + 1], R & 3,  C & 3);
    cplx e = cmul(s0, s1);
    int off = blockIdx.x * 256 + t;       // (b*4+pr)*256 + t
    rxRe[off] = e.re; rxIm[off] = e.im;
}

__device__ inline cplx rot_elem(float phi, float th, float om, int i, int j)
{
    float c = cosf(0.5f * th), s = sinf(0.5f * th);
    float mag, ph;
    if (i == 0 && j == 0)      { mag =  c; ph = -0.5f * (om + phi); }
    else if (i == 0 && j == 1) { mag = -s; ph = -0.5f * (om - phi); }
    else if (i == 1 && j == 0) { mag =  s; ph =  0.5f * (om - phi); }
    else                       { mag =  c; ph =  0.5f * (om + phi); }
    return {mag * cosf(ph), mag * sinf(ph)};
}
__device__ inline cplx superop_rot(float phi, float th, float om, int g, int gp)
{
    int r = g >> 1, c = g & 1, rp = gp >> 1, cp = gp & 1;
    return cmul(rot_elem(phi, th, om, r, rp), cconj(rot_elem(phi, th, om, c, cp)));
}

__global__ void k_build_rot(const float* __restrict__ w, float* rotRe, float* rotIm)
{
    const int l  = blockIdx.x >> 2;
    const int pr = blockIdx.x & 3;
    const int t  = threadIdx.x;
    const float* a0 = w + (l * QN + 2 * pr + 0) * 3;
    const float* a1 = w + (l * QN + 2 * pr + 1) * 3;
    const int R = t >> 4, C = t & 15;
    cplx s0 = superop_rot(a0[0], a0[1], a0[2], R >> 2, C >> 2);
    cplx s1 = superop_rot(a1[0], a1[1], a1[2], R & 3,  C & 3);
    cplx e = cmul(s0, s1);
    int off = blockIdx.x * 256 + t;
    rotRe[off] = e.re; rotIm[off] = e.im;
}

// ---------------------------------------------------------------------------
// State init: vec(|0..0><0..0|)
// ---------------------------------------------------------------------------
__global__ void k_init(float* sRe, float* sIm)
{
    const size_t total = (size_t)QB * DIM;
    for (size_t i = (size_t)blockIdx.x * blockDim.x + threadIdx.x; i < total;
         i += (size_t)gridDim.x * blockDim.x) {
        sRe[i] = ((i & (DIM - 1)) == 0) ? 1.0f : 0.0f;
        sIm[i] = 0.0f;
    }
}

// ---------------------------------------------------------------------------
// Core: apply 16x16 complex superop M to the pair of 4-level axes (qa, qb),
// in place, via v_wmma_f32_16x16x4_f32 (4 real WMMAs per complex k-step).
// Each wave owns complete fibers -> reads strictly before writes, no hazard.
// Grid: QB*2 blocks of 256 threads; batch index b derives from blockIdx only
// so the state/matrix base pointers stay in SGPRs (saddr-form vmem).
// ---------------------------------------------------------------------------
__global__ void __launch_bounds__(256)
k_apply16(float* __restrict__ sRe, float* __restrict__ sIm,
          const float* __restrict__ mRe, const float* __restrict__ mIm,
          int matBatchStride, int qa, int qb)
{
    const int lane      = threadIdx.x & 31;
    const int waveInBlk = threadIdx.x >> 5;               // 0..7
    const int b         = blockIdx.x >> 1;                // uniform (SGPR)
    const int tcBase    = ((((blockIdx.x & 1) << 3) + waveInBlk) << 4); // 16 cols/wave

    const int sha = 2 * (7 - qa);
    const int shb = 2 * (7 - qb);
    const int p1 = (sha < shb) ? sha : shb;
    const int p2 = (sha < shb) ? shb : sha;
    const unsigned lowMask = (1u << p1) - 1u;
    const int midBits = p2 - p1 - 2;
    const unsigned midMask = (1u << midBits) - 1u;

    // A fragments of M: element (m = lane%16, k = kc*4 + 2*(lane/16) + v)
    const float* __restrict__ MR = mRe + (size_t)b * matBatchStride;  // uniform base
    const float* __restrict__ MI = mIm + (size_t)b * matBatchStride;
    const int m   = lane & 15;
    const int khi = (lane >> 4) << 1;             // 0 or 2
    v2f aRe[4], aIm[4], aImN[4];
    for (int kc = 0; kc < 4; ++kc) {
        int k0 = kc * 4 + khi;
        aRe[kc][0]  = MR[m * 16 + k0];     aRe[kc][1]  = MR[m * 16 + k0 + 1];
        aIm[kc][0]  = MI[m * 16 + k0];     aIm[kc][1]  = MI[m * 16 + k0 + 1];
        aImN[kc][0] = -aIm[kc][0];         aImN[kc][1] = -aIm[kc][1];
    }

    float* __restrict__ base_re = sRe + (size_t)b * DIM;  // uniform base (SGPR)
    float* __restrict__ base_im = sIm + (size_t)b * DIM;
    const int n = lane & 15;                      // output/input column within tile

    // per-lane k-dependent index bits, constant across tiles
    unsigned koff[4][2];
    for (int kc = 0; kc < 4; ++kc)
        for (int v = 0; v < 2; ++v) {
            int k = kc * 4 + khi + v;
            koff[kc][v] = ((unsigned)(k >> 2) << sha) | ((unsigned)(k & 3) << shb);
        }

    for (int tci = 0; tci < 16; ++tci) {
        const int tc = tcBase + tci;
        const unsigned rest = (unsigned)(tc * 16 + n);   // 12-bit "rest" index
        const unsigned dep = (rest & lowMask)
                           | (((rest >> p1) & midMask) << (p1 + 2))
                           | ((rest >> (p1 + midBits)) << (p2 + 2));

        v8f accRe = {}; v8f accIm = {};
        for (int kc = 0; kc < 4; ++kc) {
            v2f bRe, bIm;
            for (int v = 0; v < 2; ++v) {
                unsigned idx = dep | koff[kc][v];
                bRe[v] = base_re[idx];
                bIm[v] = base_im[idx];
            }
            // complex D = M * S :  Dre += Mre*Sre - Mim*Sim ; Dim += Mre*Sim + Mim*Sre
            accRe = __builtin_amdgcn_wmma_f32_16x16x4_f32(false, aRe[kc],  false, bRe, (short)0, accRe, false, false);
            accRe = __builtin_amdgcn_wmma_f32_16x16x4_f32(false, aImN[kc], false, bIm, (short)0, accRe, false, false);
            accIm = __builtin_amdgcn_wmma_f32_16x16x4_f32(false, aRe[kc],  false, bIm, (short)0, accIm, false, false);
            accIm = __builtin_amdgcn_wmma_f32_16x16x4_f32(false, aIm[kc],  false, bRe, (short)0, accIm, false, false);
        }

        // D layout: VGPR d holds row (d + 8*(lane/16)), col = lane%16
        const int mhi = (lane >> 4) << 3;
        for (int d = 0; d < 8; ++d) {
            int g = d + mhi;
            unsigned idx = dep | ((unsigned)(g >> 2) << sha)
                               | ((unsigned)(g & 3) << shb);
            base_re[idx] = accRe[d];
            base_im[idx] = accIm[d];
        }
    }
}

// ---------------------------------------------------------------------------
// Diagonal -> per-qubit <Z>, shot noise, clip
// ---------------------------------------------------------------------------
__global__ void k_expect(const float* __restrict__ sRe,
                         const float* __restrict__ noise,
                         float* __restrict__ out)
{
    __shared__ float acc[QN];
    const int b = blockIdx.x;
    const int t = threadIdx.x;                  // 256 diag entries
    if (t < QN) acc[t] = 0.0f;
    __syncthreads();

    unsigned idx = 0;                            // spread bits: diag entry
    for (int j = 0; j < QN; ++j)
        if ((t >> j) & 1) idx |= (3u << (2 * j));
    float pr = sRe[(size_t)b * DIM + idx];

    for (int w = 0; w < QN; ++w) {
        float s = ((t >> (7 - w)) & 1) ? -pr : pr;
        atomicAdd(&acc[w], s);
    }
    __syncthreads();

    if (t < QN) {
        float e = acc[t];
        float sig = sqrtf(fmaxf(1.0f - e * e, 0.0f) / SHOTS_F);
        float o = e + noise[b * QN + t] * sig;
        out[b * QN + t] = fminf(1.0f, fmaxf(-1.0f, o));
    }
}

// ---------------------------------------------------------------------------
extern "C" void kernel_launch(void* const* d_in, const int* in_sizes, int n_in,
                              void* d_out, int out_size, void* d_ws, size_t ws_size,
                              hipStream_t stream)
{
    const float* x       = (const float*)d_in[0];   // (B, N)
    const float* weights = (const float*)d_in[1];   // (L, N, 3)
    const float* noise   = (const float*)d_in[2];   // (B, N)
    float* out = (float*)d_out;                     // (B, N)

    float* ws = (float*)d_ws;
    size_t o = 0;
    float* sRe   = ws + o; o += (size_t)QB * DIM;
    float* sIm   = ws + o; o += (size_t)QB * DIM;
    float* n16Re = ws + o; o += 256;
    float* n16Im = ws + o; o += 256;
    float* cnRe  = ws + o; o += 256;
    float* cnIm  = ws + o; o += 256;
    float* rxRe  = ws + o; o += (size_t)QB * 4 * 256;
    float* rxIm  = ws + o; o += (size_t)QB * 4 * 256;
    float* rotRe = ws + o; o += (size_t)QL * 4 * 256;
    float* rotIm = ws + o; o += (size_t)QL * 4 * 256;

    k_build_const<<<1, 256, 0, stream>>>(n16Re, n16Im, cnRe, cnIm);
    k_build_rx<<<QB * 4, 256, 0, stream>>>(x, rxRe, rxIm);
    k_build_rot<<<QL * 4, 256, 0, stream>>>(weights, rotRe, rotIm);
    k_init<<<4096, 256, 0, stream>>>(sRe, sIm);

    const dim3 blk(256);
    const dim3 grd(QB * 2);        // QB*16 waves, 8 waves/block, b uniform/block

    // RX layer (batch-dependent pair matrices)
    for (int p = 0; p < 4; ++p)
        k_apply16<<<grd, blk, 0, stream>>>(sRe, sIm, rxRe + p * 256, rxIm + p * 256,
                                           4 * 256, 2 * p, 2 * p + 1);
    // noise (composed channel pair matrix, shared)
    for (int p = 0; p < 4; ++p)
        k_apply16<<<grd, blk, 0, stream>>>(sRe, sIm, n16Re, n16Im, 0, 2 * p, 2 * p + 1);
    // entangling layers
    for (int l = 0; l < QL; ++l) {
        for (int p = 0; p < 4; ++p)
            k_apply16<<<grd, blk, 0, stream>>>(sRe, sIm,
                                               rotRe + (l * 4 + p) * 256,
                                               rotIm + (l * 4 + p) * 256,
                                               0, 2 * p, 2 * p + 1);
        int r = l % (QN - 1) + 1;
        for (int w = 0; w < QN; ++w)
            k_apply16<<<grd, blk, 0, stream>>>(sRe, sIm, cnRe, cnIm, 0, w, (w + r) % QN);
    }
    // trailing noise
    for (int p = 0; p < 4; ++p)
        k_apply16<<<grd, blk, 0, stream>>>(sRe, sIm, n16Re, n16Im, 0, 2 * p, 2 * p + 1);

    k_expect<<<QB, 256, 0, stream>>>(sRe, noise, out);
}